// ResonanceTilingKernel_2980707303805
// MI455X (gfx1250) — compile-verified
//
#include <hip/hip_runtime.h>

// CDNA5 / gfx1250 WMMA types
typedef float v2f __attribute__((ext_vector_type(2)));
typedef float v8f __attribute__((ext_vector_type(8)));

#define WAVES_PER_WG 4
#define BLOCK_THREADS 128
#define ROW_FLOATS 1024      // 256 quat-components * 4 floats per batch row
#define BUF_STRIDE 68        // staging row stride (floats): conflict-free A/C-frag reads
#define W_PAIR_STRIDE 160    // K-pair row stride (floats): conflict-free b64 B-frag reads

// y1 = x1 + x2 @ Wphi[t] ; y2 = x2 + y1 @ Wpsi[8+t]   (per 16-quat tile t, 64x64 real matmuls)
__global__ __launch_bounds__(BLOCK_THREADS)
void hamilton_wmma_kernel(const float* __restrict__ x,
                          const float* __restrict__ phi,   // [16][16][16][4] f32
                          const float* __restrict__ psi,   // [16][16][16][4] f32
                          float* __restrict__ y,
                          int B)
{
    // W stored K-pair interleaved: element (row,col) -> [ (row>>1)*160 + 2*col + (row&1) ]
    // so a B fragment (rows klo,klo+1 ; one col) is ONE aligned ds_load_b64.
    __shared__ float Wphi[32 * W_PAIR_STRIDE];                  // 20480 B
    __shared__ float Wpsi[32 * W_PAIR_STRIDE];                  // 20480 B
    __shared__ float Buf[WAVES_PER_WG][16 * BUF_STRIDE];        // 17408 B (per-wave staging)

    const int tid  = threadIdx.x;
    const int wave = tid >> 5;
    const int lane = tid & 31;
    const int t    = blockIdx.x;            // tile 0..7

    // ---------------- expand Hamilton right-mult matrices into LDS ----------------
    // W[i*4+ci][j*4+co] = sign(ci,co) * w[t][i][j][ci^co]; neg-bit mask over (ci*4+co) = 0x3950
    const float* __restrict__ wsrc0 = phi + (size_t)t * 1024;
    const float* __restrict__ wsrc1 = psi + (size_t)(8 + t) * 1024;
    for (int idx = tid; idx < 2 * 64 * 64; idx += BLOCK_THREADS) {
        const int m   = idx >> 12;          // 0 = phi, 1 = psi
        const int e   = idx & 4095;
        const int row = e >> 6;             // K index 0..63
        const int col = e & 63;             // N index 0..63
        const int i = row >> 2, ci = row & 3;
        const int j = col >> 2, co = col & 3;
        const float raw = (m ? wsrc1 : wsrc0)[i * 64 + j * 4 + (ci ^ co)];
        const float val = ((0x3950u >> (ci * 4 + co)) & 1u) ? -raw : raw;
        (m ? Wpsi : Wphi)[(row >> 1) * W_PAIR_STRIDE + (col << 1) + (row & 1)] = val;
    }
    __syncthreads();

    // ---------------- per-lane fragment coordinates ----------------
    const int row2  = lane >> 1;            // staging row this lane loads/stores (0..15)
    const int half  = lane & 1;             // which 128B half of the 256B row slice
    const int Nc    = lane & 15;            // C/D + B fragment column
    const int Moff  = (lane >> 4) << 3;     // C/D lanes 16..31 hold M = v + 8
    const int klo   = (lane >> 4) << 1;     // A lanes 16..31 hold K = {2,3}
    const int khalf = lane >> 4;            // B K-pair select (pair index 2*kb + khalf)

    float* __restrict__ buf = &Buf[wave][0];

    const int nChunks    = B >> 4;                       // 16 batch rows per chunk
    const int waveGlobal = blockIdx.y * WAVES_PER_WG + wave;
    const int waveStride = gridDim.y * WAVES_PER_WG;

    for (int g = waveGlobal; g < nChunks; g += waveStride) {
        const size_t rowBase = (size_t)g * 16;
        const float* __restrict__ x1p = x + rowBase * ROW_FLOATS + t * 64;   // x1 tile slice
        const float* __restrict__ x2p = x1p + 512;                           // x2 tile slice
        float* __restrict__ y1p = y + rowBase * ROW_FLOATS + t * 64;
        float* __restrict__ y2p = y1p + 512;

        // prefetch next chunk (wave-uniform branch; lowers to global_prefetch_b8)
        if (g + waveStride < nChunks) {
            const float* nxt = x + ((size_t)(g + waveStride) * 16 + row2) * ROW_FLOATS + t * 64;
            __builtin_prefetch(nxt + half * 32, 0, 1);
            __builtin_prefetch(nxt + 512 + half * 32, 0, 1);
        }

        // ---- stage x1 (each lane: one contiguous 128B half-row, 8x b128) ----
        {
            const float4* __restrict__ src = (const float4*)(x1p + (size_t)row2 * ROW_FLOATS + half * 32);
            float4* __restrict__ dst = (float4*)(buf + row2 * BUF_STRIDE + half * 32);
            #pragma unroll
            for (int k = 0; k < 8; ++k) dst[k] = src[k];
        }
        // x1 in C/D layout -> WMMA accumulator init (residual add for free)
        v8f y1c[4];
        #pragma unroll
        for (int nb = 0; nb < 4; ++nb) {
            #pragma unroll
            for (int v = 0; v < 8; ++v)
                y1c[nb][v] = buf[(v + Moff) * BUF_STRIDE + nb * 16 + Nc];
        }

        // ---- stage x2 (overwrites x1 staging), read A-frags + C-frags ----
        {
            const float4* __restrict__ src = (const float4*)(x2p + (size_t)row2 * ROW_FLOATS + half * 32);
            float4* __restrict__ dst = (float4*)(buf + row2 * BUF_STRIDE + half * 32);
            #pragma unroll
            for (int k = 0; k < 8; ++k) dst[k] = src[k];
        }
        v2f a[16];
        #pragma unroll
        for (int kb = 0; kb < 16; ++kb)
            a[kb] = *(const v2f*)(buf + (lane & 15) * BUF_STRIDE + kb * 4 + klo);
        v8f x2c[4];
        #pragma unroll
        for (int nb = 0; nb < 4; ++nb) {
            #pragma unroll
            for (int v = 0; v < 8; ++v)
                x2c[nb][v] = buf[(v + Moff) * BUF_STRIDE + nb * 16 + Nc];
        }

        // ---- GEMM1: y1 = x1 + x2 @ Wphi ----
        #pragma unroll
        for (int nb = 0; nb < 4; ++nb) {
            v8f acc = y1c[nb];
            #pragma unroll
            for (int kb = 0; kb < 16; ++kb) {
                const v2f b = *(const v2f*)(Wphi + (2 * kb + khalf) * W_PAIR_STRIDE
                                                 + ((nb * 16 + Nc) << 1));
                acc = __builtin_amdgcn_wmma_f32_16x16x4_f32(
                          false, a[kb], false, b, (short)0, acc, false, false);
            }
            y1c[nb] = acc;
        }

        // ---- restage y1 into buf (C-layout -> row-major), reload A-frags ----
        #pragma unroll
        for (int nb = 0; nb < 4; ++nb) {
            #pragma unroll
            for (int v = 0; v < 8; ++v)
                buf[(v + Moff) * BUF_STRIDE + nb * 16 + Nc] = y1c[nb][v];
        }
        #pragma unroll
        for (int kb = 0; kb < 16; ++kb)
            a[kb] = *(const v2f*)(buf + (lane & 15) * BUF_STRIDE + kb * 4 + klo);

        // ---- y1 copy-out from staging: contiguous 128B per lane, b128 stores ----
        {
            const float4* __restrict__ src = (const float4*)(buf + row2 * BUF_STRIDE + half * 32);
            float4* __restrict__ dst = (float4*)(y1p + (size_t)row2 * ROW_FLOATS + half * 32);
            #pragma unroll
            for (int k = 0; k < 8; ++k) dst[k] = src[k];
        }

        // ---- GEMM2: y2 = x2 + y1 @ Wpsi ----
        v8f y2c[4];
        #pragma unroll
        for (int nb = 0; nb < 4; ++nb) {
            v8f acc = x2c[nb];
            #pragma unroll
            for (int kb = 0; kb < 16; ++kb) {
                const v2f b = *(const v2f*)(Wpsi + (2 * kb + khalf) * W_PAIR_STRIDE
                                                 + ((nb * 16 + Nc) << 1));
                acc = __builtin_amdgcn_wmma_f32_16x16x4_f32(
                          false, a[kb], false, b, (short)0, acc, false, false);
            }
            y2c[nb] = acc;
        }

        // ---- restage y2 and copy out with b128 stores ----
        #pragma unroll
        for (int nb = 0; nb < 4; ++nb) {
            #pragma unroll
            for (int v = 0; v < 8; ++v)
                buf[(v + Moff) * BUF_STRIDE + nb * 16 + Nc] = y2c[nb][v];
        }
        {
            const float4* __restrict__ src = (const float4*)(buf + row2 * BUF_STRIDE + half * 32);
            float4* __restrict__ dst = (float4*)(y2p + (size_t)row2 * ROW_FLOATS + half * 32);
            #pragma unroll
            for (int k = 0; k < 8; ++k) dst[k] = src[k];
        }
    }
}

extern "C" void kernel_launch(void* const* d_in, const int* in_sizes, int n_in,
                              void* d_out, int out_size, void* d_ws, size_t ws_size,
                              hipStream_t stream) {
    const float* x   = (const float*)d_in[0];
    const float* phi = (const float*)d_in[1];
    const float* psi = (const float*)d_in[2];
    float* yo = (float*)d_out;

    const int B = in_sizes[0] / ROW_FLOATS;   // 65536
    const int nChunks = B >> 4;

    int gy = 256;                             // 4 chunks per wave at B=65536
    const int maxgy = (nChunks + WAVES_PER_WG - 1) / WAVES_PER_WG;
    if (gy > maxgy) gy = maxgy;
    if (gy < 1) gy = 1;

    dim3 grid(8, gy);
    hamilton_wmma_kernel<<<grid, BLOCK_THREADS, 0, stream>>>(x, phi, psi, yo, B);
}